// PSRLoss_32040456028692
// MI455X (gfx1250) — compile-verified
//
#include <hip/hip_runtime.h>
#include <math.h>

// ---------------- problem constants (reference hard-codes these) ----------
#define HH 2048
#define WW 2048
#define NB 8
#define NPB (HH * WW)            // 4,194,304 elements per batch
#define NV4 (NPB / 4)            // 1,048,576 float4 per batch
#define BLK 256
#define BPB 512                  // blocks per batch
#define PER_THREAD (NV4 / (BPB * BLK))   // = 8, exact -> no divergence

typedef __attribute__((ext_vector_type(2))) float v2f;
typedef __attribute__((ext_vector_type(4))) float f32x4;
typedef __attribute__((ext_vector_type(8))) float v8f;

struct WS {
    unsigned long long pmax[NB]; // (ordered(max)<<32) | ~idx
    float sum[NB];
    float sq[NB];
    float cnt[NB];
};

// monotone float -> uint mapping so integer max == float max
__device__ __forceinline__ unsigned long long packMax(float v, unsigned idx) {
    unsigned b = __float_as_uint(v);
    unsigned ord = (b & 0x80000000u) ? ~b : (b | 0x80000000u);
    return ((unsigned long long)ord << 32) | (unsigned long long)(~idx);
}

__global__ void psr_init_ws(WS* __restrict__ ws) {
    int t = threadIdx.x;
    if (t < NB) {
        ws->pmax[t] = 0ull;
        ws->sum[t]  = 0.0f;
        ws->sq[t]   = 0.0f;
        ws->cnt[t]  = 0.0f;
    }
}

__global__ __launch_bounds__(BLK) void psr_reduce(const float* __restrict__ resp,
                                                  WS* __restrict__ ws) {
    const int b = blockIdx.y;
    const f32x4* __restrict__ base =
        (const f32x4*)(resp + (size_t)b * (size_t)NPB);

    const unsigned stride = BPB * BLK;
    unsigned i = blockIdx.x * BLK + threadIdx.x;

    float s = 0.0f, q = 0.0f, c = 0.0f;
    float mx = -3.0e38f;
    unsigned mi = 0u;

#pragma unroll
    for (int it = 0; it < PER_THREAD; ++it, i += stride) {
        // read-once streaming data: non-temporal b128 load
        f32x4 v = __builtin_nontemporal_load(base + i);
#pragma unroll
        for (int k = 0; k < 4; ++k) {
            float x = v[k];
            bool p = x > 0.0f;
            c += p ? 1.0f : 0.0f;
            s += p ? x : 0.0f;
            q += p ? x * x : 0.0f;
            unsigned idx = i * 4u + (unsigned)k;
            if (x > mx) { mx = x; mi = idx; }
        }
    }

    const int lane = threadIdx.x & 31;
    const int wv   = threadIdx.x >> 5;

    // ---- wave reduce max/argmax + count via shuffles (not additive / small) ----
    unsigned long long pk = packMax(mx, mi);
#pragma unroll
    for (int off = 16; off > 0; off >>= 1) {
        unsigned long long o = __shfl_down(pk, (unsigned)off, 32);
        if (o > pk) pk = o;
        c += __shfl_down(c, (unsigned)off, 32);
    }

    // ---- wave reduce (sum, sumsq) with V_WMMA_F32_16X16X4_F32 ----
    // Stage 1: A[m] = [s_m, q_m, s_{m+16}, q_{m+16}] (A 16x4 layout: lanes 0-15
    // supply K=0,1; lanes 16-31 supply K=2,3). B selects columns: cols 0-7 take
    // K=0,2 (sums), cols 8-15 take K=1,3 (squares).
    v2f a;    a.x = s; a.y = q;
    v2f bsel; bsel.x = ((lane & 15) < 8) ? 1.0f : 0.0f;
              bsel.y = ((lane & 15) < 8) ? 0.0f : 1.0f;
    v8f zero = {};
    v8f d1 = __builtin_amdgcn_wmma_f32_16x16x4_f32(
        false, a, false, bsel, (short)0, zero, false, false);
    // d1[v] lanes 0-15 hold D1[v][n] = s_v+s_{v+16} (n<8) / q_v+q_{v+16} (n>=8);
    // lanes 16-31 hold rows v+8.

    // Stage 2: fold the 16 rows with A = ones, feeding D1 VGPR pairs back as B
    // (B 4x16 f32 layout: vgpr0 -> K0/K2, vgpr1 -> K1/K3 by lane half), chained
    // through the accumulator. Each column keeps its identity, so after 4 WMMAs
    // every element of column n holds total_sum (n<8) or total_sumsq (n>=8).
    v2f ones; ones.x = 1.0f; ones.y = 1.0f;
    v8f acc = {};
#pragma unroll
    for (int j = 0; j < 4; ++j) {
        v2f b2; b2.x = d1[2 * j]; b2.y = d1[2 * j + 1];
        acc = __builtin_amdgcn_wmma_f32_16x16x4_f32(
            false, ones, false, b2, (short)0, acc, false, false);
    }
    float stot = __shfl(acc[0], 0, 32);  // D[0][0]  = total wave sum
    float qtot = __shfl(acc[0], 8, 32);  // D[0][8]  = total wave sumsq

    // ---- cross-wave (8 waves) via LDS, then one atomic set per block ----
    __shared__ float lsum[BLK / 32], lsq[BLK / 32], lcnt[BLK / 32];
    __shared__ unsigned long long lmax[BLK / 32];
    if (lane == 0) { lsum[wv] = stot; lsq[wv] = qtot; lcnt[wv] = c; lmax[wv] = pk; }
    __syncthreads();
    if (threadIdx.x == 0) {
        float S = 0.0f, Q = 0.0f, C = 0.0f;
        unsigned long long P = 0ull;
#pragma unroll
        for (int j = 0; j < BLK / 32; ++j) {
            S += lsum[j]; Q += lsq[j]; C += lcnt[j];
            if (lmax[j] > P) P = lmax[j];
        }
        atomicAdd(&ws->sum[b], S);
        atomicAdd(&ws->sq[b],  Q);
        atomicAdd(&ws->cnt[b], C);
        atomicMax(&ws->pmax[b], P);
    }
}

__global__ __launch_bounds__(1024) void psr_finalize(const float* __restrict__ resp,
                                                     WS* __restrict__ ws,
                                                     float* __restrict__ out) {
    __shared__ float wsum[NB], wsq[NB], wcnt[NB];
    __shared__ float psr[NB];
    const int t = threadIdx.x;
    if (t < NB) { wsum[t] = 0.0f; wsq[t] = 0.0f; wcnt[t] = 0.0f; }
    __syncthreads();

    // subtract 11x11 window contributions (<=121 elems per batch)
    if (t < NB * 121) {
        int b = t / 121, i = t % 121;
        int dy = i / 11 - 5, dx = i % 11 - 5;
        unsigned long long P = ws->pmax[b];
        unsigned idx = ~(unsigned)(P & 0xFFFFFFFFull);
        int cy = (int)(idx / WW), cx = (int)(idx % WW);
        int y = cy + dy, x = cx + dx;
        if (y >= 0 && y < HH && x >= 0 && x < WW) {
            float v = resp[(size_t)b * (size_t)NPB + (size_t)y * WW + x];
            if (v > 0.0f) {
                atomicAdd(&wsum[b], v);
                atomicAdd(&wsq[b],  v * v);
                atomicAdd(&wcnt[b], 1.0f);
            }
        }
    }
    __syncthreads();

    if (t < NB) {
        unsigned long long P = ws->pmax[t];
        unsigned ord  = (unsigned)(P >> 32);
        unsigned bits = (ord & 0x80000000u) ? (ord ^ 0x80000000u) : ~ord;
        float fmax = __uint_as_float(bits);
        double n = (double)(ws->cnt[t] - wcnt[t]);
        double s = (double)ws->sum[t] - (double)wsum[t];
        double q = (double)ws->sq[t]  - (double)wsq[t];
        double mean = s / n;
        double var  = (q - s * s / n) / (n - 1.0);   // unbiased, ddof=1
        double sd   = sqrt(var);
        psr[t] = (float)(((double)fmax - mean) / (sd + 1e-5));
    }
    __syncthreads();
    if (t == 0) {
        float m = 0.0f;
#pragma unroll
        for (int j = 0; j < NB; ++j) m += psr[j];
        out[0] = m / (float)NB;
    }
}

extern "C" void kernel_launch(void* const* d_in, const int* in_sizes, int n_in,
                              void* d_out, int out_size, void* d_ws, size_t ws_size,
                              hipStream_t stream) {
    (void)in_sizes; (void)n_in; (void)out_size; (void)ws_size;
    const float* resp = (const float*)d_in[0];
    float* out = (float*)d_out;
    WS* ws = (WS*)d_ws;

    psr_init_ws<<<1, 32, 0, stream>>>(ws);
    psr_reduce<<<dim3(BPB, NB, 1), BLK, 0, stream>>>(resp, ws);
    psr_finalize<<<1, 1024, 0, stream>>>(resp, ws, out);
}